// NextState_8469675508286
// MI455X (gfx1250) — compile-verified
//
#include <hip/hip_runtime.h>
#include <cstdint>

#define BLK 256
#define DTC 0.1f
#define EPSQ 1e-12f

// ---- gfx1250 async global<->LDS helpers (ASYNCcnt-tracked, CDNA5 ISA §10 async ops) ----
__device__ __forceinline__ void async_load_b128(uint32_t lds_off, const void* gaddr) {
    asm volatile("global_load_async_to_lds_b128 %0, %1, off"
                 :: "v"(lds_off), "v"((uint64_t)(uintptr_t)gaddr) : "memory");
}
__device__ __forceinline__ void async_load_b32(uint32_t lds_off, const void* gaddr) {
    asm volatile("global_load_async_to_lds_b32 %0, %1, off"
                 :: "v"(lds_off), "v"((uint64_t)(uintptr_t)gaddr) : "memory");
}
__device__ __forceinline__ void async_store_b128(const void* gaddr, uint32_t lds_off) {
    asm volatile("global_store_async_from_lds_b128 %0, %1, off"
                 :: "v"((uint64_t)(uintptr_t)gaddr), "v"(lds_off) : "memory");
}
__device__ __forceinline__ void async_store_b32(const void* gaddr, uint32_t lds_off) {
    asm volatile("global_store_async_from_lds_b32 %0, %1, off"
                 :: "v"((uint64_t)(uintptr_t)gaddr), "v"(lds_off) : "memory");
}
__device__ __forceinline__ void wait_asynccnt0() {
    asm volatile("s_wait_asynccnt 0" ::: "memory");
}

__global__ __launch_bounds__(BLK) void next_state_kernel(
    const float* __restrict__ state,   // [K,13]
    const float* __restrict__ delta,   // [K,6]
    float* __restrict__ out,           // [K,13]
    int K)
{
    __shared__ __align__(16) float sState[BLK * 13];  // 13312 B: input slab, reused for output
    __shared__ __align__(16) float sDelta[BLK * 6];   //  6144 B

    const int tid  = threadIdx.x;
    const int base = blockIdx.x * BLK;
    const int n    = min(BLK, K - base);
    if (n <= 0) return;

    const int ndS = n * 13;           // dwords in state slab
    const int ndD = n * 6;            // dwords in delta slab
    const float* gS = state + (size_t)base * 13;
    const float* gD = delta + (size_t)base * 6;
    float*       gO = out   + (size_t)base * 13;

    const uint32_t sStateOff = (uint32_t)(uintptr_t)(&sState[0]); // low 32b of flat LDS addr = LDS offset
    const uint32_t sDeltaOff = (uint32_t)(uintptr_t)(&sDelta[0]);

    // ---- Stage 1: async coalesced B128 copy global -> LDS (flat dword-linear layout) ----
    const int nv4S = ndS >> 2;
    for (int j = tid; j < nv4S; j += BLK)
        async_load_b128(sStateOff + (uint32_t)j * 16u, gS + (size_t)j * 4);
    for (int j = (nv4S << 2) + tid; j < ndS; j += BLK)   // remainder guard (empty for n%4==0)
        async_load_b32(sStateOff + (uint32_t)j * 4u, gS + j);

    const int nv4D = ndD >> 2;
    for (int j = tid; j < nv4D; j += BLK)
        async_load_b128(sDeltaOff + (uint32_t)j * 16u, gD + (size_t)j * 4);
    for (int j = (nv4D << 2) + tid; j < ndD; j += BLK)
        async_load_b32(sDeltaOff + (uint32_t)j * 4u, gD + j);

    wait_asynccnt0();
    __syncthreads();

    // ---- Stage 2: per-entity rigid-body update from LDS (stride-13 => conflict-free banks) ----
    if (tid < n) {
        const float* s = &sState[tid * 13];
        float px = s[0], py = s[1], pz = s[2];
        float x  = s[3], y  = s[4], z  = s[5], w = s[6];
        float vx = s[7], vy = s[8], vz = s[9];
        float ox = s[10], oy = s[11], oz = s[12];
        const float* d = &sDelta[tid * 6];
        float d0 = d[0], d1 = d[1], d2 = d[2], d3 = d[3], d4 = d[4], d5 = d[5];

        // posDot = R(q) * v
        float pdx = (1.f - 2.f*(y*y + z*z))*vx + 2.f*(x*y - z*w)*vy + 2.f*(x*z + y*w)*vz;
        float pdy = 2.f*(x*y + z*w)*vx + (1.f - 2.f*(x*x + z*z))*vy + 2.f*(y*z - x*w)*vz;
        float pdz = 2.f*(x*z - y*w)*vx + 2.f*(y*z + x*w)*vy + (1.f - 2.f*(x*x + y*y))*vz;

        // quatDot = 0.5 * T(q) * omega
        float qd0 = 0.5f * (-x*ox - y*oy - z*oz);
        float qd1 = 0.5f * ( w*ox - z*oy + y*oz);
        float qd2 = 0.5f * ( z*ox + w*oy - x*oz);
        float qd3 = 0.5f * (-y*ox + x*oy + w*oz);

        px += DTC * pdx; py += DTC * pdy; pz += DTC * pdz;
        float nx = x + DTC * qd0, ny = y + DTC * qd1, nz = z + DTC * qd2, nw = w + DTC * qd3;
        float sq = nx*nx + ny*ny + nz*nz + nw*nw;
        float rs = rsqrtf(fmaxf(sq, EPSQ));
        nx *= rs; ny *= rs; nz *= rs; nw *= rs;

        float* o = &sState[tid * 13];        // reuse own slot: no cross-thread hazard
        o[0] = px;      o[1] = py;      o[2] = pz;
        o[3] = nx;      o[4] = ny;      o[5] = nz;      o[6] = nw;
        o[7] = vx + d0; o[8] = vy + d1; o[9] = vz + d2;
        o[10] = ox + d3; o[11] = oy + d4; o[12] = oz + d5;
    }
    __syncthreads();

    // ---- Stage 3: async coalesced B128 drain LDS -> global ----
    for (int j = tid; j < nv4S; j += BLK)
        async_store_b128(gO + (size_t)j * 4, sStateOff + (uint32_t)j * 16u);
    for (int j = (nv4S << 2) + tid; j < ndS; j += BLK)
        async_store_b32(gO + j, sStateOff + (uint32_t)j * 4u);

    wait_asynccnt0();   // S_ENDPGM also implies wait-idle; explicit for clarity
}

extern "C" void kernel_launch(void* const* d_in, const int* in_sizes, int n_in,
                              void* d_out, int out_size, void* d_ws, size_t ws_size,
                              hipStream_t stream) {
    const float* state = (const float*)d_in[0];
    const float* delta = (const float*)d_in[1];
    float*       out   = (float*)d_out;
    const int K = in_sizes[0] / 13;
    const int grid = (K + BLK - 1) / BLK;
    next_state_kernel<<<grid, BLK, 0, stream>>>(state, delta, out, K);
}